// FirstStageRetriever_83519934038683
// MI455X (gfx1250) — compile-verified
//
#include <hip/hip_runtime.h>

typedef float v2f __attribute__((ext_vector_type(2)));
typedef float v8f __attribute__((ext_vector_type(8)));

#define NQ   16
#define DIM  384
#define ZSCALE -10.0f      // -1/REG, REG = 0.1, DESCENDING
#define EPSN 1e-12f

#define CHNK   2048        // LDS-resident bitonic chunk (16KB of u64 keys)
#define SORT_T 256

#if defined(__has_builtin)
#if __has_builtin(__builtin_amdgcn_global_load_async_to_lds_b64)
#define HAVE_ASYNC_LDS 1
#endif
#endif

#ifdef HAVE_ASYNC_LDS
typedef int v2i_ __attribute__((vector_size(2 * sizeof(int))));
typedef __attribute__((address_space(1))) v2i_* glb_v2ip;   // prints as "__device__ *"
typedef __attribute__((address_space(3))) v2i_* lds_v2ip;   // prints as "__shared__ *"
#endif

// ---------------------------------------------------------------------------
// Cooperative global -> LDS fill. Uses CDNA5 async LDS loads when available
// (GLOBAL_LOAD_ASYNC_TO_LDS_B64, tracked by ASYNCcnt), else plain ld/st.
// Caller must __syncthreads() afterwards.
// ---------------------------------------------------------------------------
__device__ __forceinline__ void lds_fill(unsigned long long* sh,
                                         const unsigned long long* g,
                                         int count, int tid, int nthr) {
#ifdef HAVE_ASYNC_LDS
  for (int t = tid; t < count; t += nthr) {
    __builtin_amdgcn_global_load_async_to_lds_b64((glb_v2ip)(g + t),
                                                  (lds_v2ip)(sh + t), 0, 0);
  }
  asm volatile("s_wait_asynccnt 0x0" ::: "memory");
#else
  for (int t = tid; t < count; t += nthr) sh[t] = g[t];
#endif
}

// ---------------------------------------------------------------------------
// 1) Normalize the 16 query rows: qn = q / max(||q||, 1e-12)
// ---------------------------------------------------------------------------
__global__ void qnorm_kernel(const float* __restrict__ q, float* __restrict__ qn) {
  __shared__ float red[128];
  const int row = blockIdx.x;
  const int t = threadIdx.x;
  float ss = 0.f;
  for (int k = t; k < DIM; k += 128) { float v = q[row * DIM + k]; ss += v * v; }
  red[t] = ss;
  __syncthreads();
  for (int off = 64; off > 0; off >>= 1) {
    if (t < off) red[t] += red[t + off];
    __syncthreads();
  }
  const float rn = 1.0f / fmaxf(sqrtf(red[0]), EPSN);
  for (int k = t; k < DIM; k += 128) qn[row * DIM + k] = q[row * DIM + k] * rn;
}

// ---------------------------------------------------------------------------
// 2) Cosine similarity GEMM with V_WMMA_F32_16X16X4_F32 (fp32 in/acc, matches
//    reference numerics; GEMM is memory-bound so fp32 costs nothing).
//    One wave per 16-doc tile; M=16 queries is exactly one WMMA tile.
// ---------------------------------------------------------------------------
__global__ void simgemm_kernel(const float* __restrict__ qn,
                               const float* __restrict__ corpus,
                               float* __restrict__ simout, int N) {
  const int wave = (int)((blockIdx.x * blockDim.x + threadIdx.x) >> 5);
  const int lane = (int)(threadIdx.x & 31);
  const int ntiles = (N + 15) >> 4;
  if (wave >= ntiles) return;                       // wave-uniform exit (EXEC all-1 for WMMA)
  const int base = wave << 4;

  const int r16  = lane & 15;                       // A row / B column / doc-in-tile
  const int koff = (lane >> 4) << 1;                // lanes 0-15 -> K{0,1}; 16-31 -> K{2,3}
  const float* aptr = qn + (size_t)r16 * DIM + koff;
  long long drow = base + r16; if (drow > (long long)N - 1) drow = N - 1;  // clamp tail loads
  const float* bptr = corpus + (size_t)drow * DIM + koff;

  v8f acc = {};
  float ss = 0.f;
  #pragma unroll 4
  for (int k0 = 0; k0 < DIM; k0 += 4) {
    v2f a = *(const v2f*)(aptr + k0);
    v2f b = *(const v2f*)(bptr + k0);
    ss += b.x * b.x + b.y * b.y;
    acc = __builtin_amdgcn_wmma_f32_16x16x4_f32(false, a, false, b,
                                                (short)0, acc, false, false);
  }
  // lane pair (l, l+16) together covers doc row (l&15): combine sum-of-squares
  const float tot = ss + __shfl_xor(ss, 16, 32);
  const float rn  = 1.0f / fmaxf(sqrtf(tot), EPSN);

  const int doc = base + r16;
  if (doc < N) {
    const int mtop = (lane >> 4) << 3;              // D layout: lanes 16-31 hold M=8..15
    #pragma unroll
    for (int r = 0; r < 8; ++r)
      simout[(long long)(mtop + r) * N + doc] = acc[r] * rn;
  }
}

// ---------------------------------------------------------------------------
// 3a) Build stable sort keys: hi32 = descending-orderable(z), lo32 = index.
//     Ascending u64 sort == stable argsort(-z). Pad to M (pow2) with max.
// ---------------------------------------------------------------------------
__global__ void key_build_kernel(const float* __restrict__ sims, int N, int M,
                                 unsigned long long* __restrict__ keys) {
  const int row = blockIdx.y;
  const int i = (int)(blockIdx.x * 256 + threadIdx.x);
  if (i >= M) return;
  unsigned long long* kr = keys + (size_t)row * M;
  if (i < N) {
    const float z = ZSCALE * sims[(size_t)row * N + i];
    unsigned u = __float_as_uint(z);
    u = (u >> 31) ? ~u : (u | 0x80000000u);     // orderable: ascending == float ascending
    const unsigned kd = ~u;                     // invert: ascending == float descending
    kr[i] = ((unsigned long long)kd << 32) | (unsigned)i;
  } else {
    kr[i] = 0xFFFFFFFFFFFFFFFFull;              // sorts past every finite key
  }
}

// ---------------------------------------------------------------------------
// 3b) LDS-resident bitonic phase covering all k = 2..klim (klim == chunk size).
//     Direction uses the GLOBAL element index so chunks alternate asc/desc
//     exactly as the full network requires.
// ---------------------------------------------------------------------------
__global__ void bitonic_chunk_kernel(unsigned long long* __restrict__ keys,
                                     int M, int klim) {
  __shared__ unsigned long long sh[CHNK];
  const int row = blockIdx.y;
  const int gbase = (int)blockIdx.x * klim;
  unsigned long long* kr = keys + (size_t)row * M + gbase;
  const int tid = (int)threadIdx.x;

  lds_fill(sh, kr, klim, tid, SORT_T);
  __syncthreads();

  for (int k = 2; k <= klim; k <<= 1) {
    for (int j = k >> 1; j > 0; j >>= 1) {
      for (int p = tid; p < (klim >> 1); p += SORT_T) {
        const int i = ((p & ~(j - 1)) << 1) | (p & (j - 1));
        const int l = i | j;
        const bool asc = (((gbase + i) & k) == 0);
        const unsigned long long a = sh[i];
        const unsigned long long b = sh[l];
        if ((a > b) == asc) { sh[i] = b; sh[l] = a; }
      }
      __syncthreads();
    }
  }
  for (int t = tid; t < klim; t += SORT_T) kr[t] = sh[t];
}

// ---------------------------------------------------------------------------
// 3c) One cross-chunk compare-exchange substage (span j >= chunk size).
//     Pair-enumerated: every thread does useful work.
// ---------------------------------------------------------------------------
__global__ void bitonic_global_kernel(unsigned long long* __restrict__ keys,
                                      int M, int k, int j) {
  const int row = blockIdx.y;
  const int p = (int)(blockIdx.x * 256 + threadIdx.x);
  if (p >= (M >> 1)) return;
  const int i = ((p & ~(j - 1)) << 1) | (p & (j - 1));
  const int l = i | j;
  unsigned long long* kr = keys + (size_t)row * M;
  const unsigned long long a = kr[i];
  const unsigned long long b = kr[l];
  if ((a > b) == ((i & k) == 0)) { kr[i] = b; kr[l] = a; }
}

// ---------------------------------------------------------------------------
// 3d) LDS-resident tail of merge phase k: substages j = CHNK/2 .. 1.
//     For k > CHNK the direction is uniform per chunk -> hoisted.
// ---------------------------------------------------------------------------
__global__ void bitonic_tail_kernel(unsigned long long* __restrict__ keys,
                                    int M, int k) {
  __shared__ unsigned long long sh[CHNK];
  const int row = blockIdx.y;
  const int gbase = (int)blockIdx.x * CHNK;
  unsigned long long* kr = keys + (size_t)row * M + gbase;
  const int tid = (int)threadIdx.x;

  lds_fill(sh, kr, CHNK, tid, SORT_T);
  __syncthreads();

  const bool asc = ((gbase & k) == 0);
  for (int j = CHNK >> 1; j > 0; j >>= 1) {
    for (int p = tid; p < (CHNK >> 1); p += SORT_T) {
      const int i = ((p & ~(j - 1)) << 1) | (p & (j - 1));
      const int l = i | j;
      const unsigned long long a = sh[i];
      const unsigned long long b = sh[l];
      if ((a > b) == asc) { sh[i] = b; sh[l] = a; }
    }
    __syncthreads();
  }
  for (int t = tid; t < CHNK; t += SORT_T) kr[t] = sh[t];
}

// ---------------------------------------------------------------------------
// 3e) Unpack: ranks[idx] = sorted position, ssorted[pos] = z value.
// ---------------------------------------------------------------------------
__global__ void unpack_kernel(const unsigned long long* __restrict__ keys,
                              const float* __restrict__ sims, int N, int M,
                              int* __restrict__ ranks, float* __restrict__ ssorted) {
  const int row = blockIdx.y;
  const int p = (int)(blockIdx.x * 256 + threadIdx.x);
  if (p >= N) return;                             // positions >= N hold padding
  const unsigned long long v = keys[(size_t)row * M + p];
  const int idx = (int)(unsigned)(v & 0xFFFFFFFFu);
  ranks[(size_t)row * N + idx] = p;
  ssorted[(size_t)row * N + p] = ZSCALE * sims[(size_t)row * N + idx];
}

// ---------------------------------------------------------------------------
// 4) Pool-Adjacent-Violators on y = s - [N..1], one serial thread per row.
//    Afterwards rewrite in place: sums[b] -> block mean, cnts[b] -> cum bound.
// ---------------------------------------------------------------------------
__global__ void pav_kernel(const float* __restrict__ ssorted, int N,
                           float* __restrict__ sums, float* __restrict__ cnts,
                           int* __restrict__ nbout) {
  if (threadIdx.x != 0) return;
  const int row = blockIdx.x;
  const float* s = ssorted + (long long)row * N;
  float* sm = sums + (long long)row * N;
  float* ct = cnts + (long long)row * N;
  int p = 0;
  for (int i = 0; i < N; ++i) {
    float cs = s[i] - (float)(N - i);    // w = N, N-1, ..., 1
    float cc = 1.0f;
    while (p > 0 && sm[p - 1] * cc < cs * ct[p - 1]) {   // prev mean < cur mean -> merge
      cs += sm[p - 1];
      cc += ct[p - 1];
      --p;
    }
    sm[p] = cs; ct[p] = cc; ++p;
  }
  float cum = 0.f;
  for (int b = 0; b < p; ++b) {
    const float c = ct[b];
    sm[b] = sm[b] / c;                   // block mean (dual value)
    cum += c;
    ct[b] = cum;                         // cumulative bound (exact: <= 50000)
  }
  nbout[row] = p;
}

// ---------------------------------------------------------------------------
// 5) out_i = z_i - dual[block(rank_i)] via binary search over bounds
//    (== searchsorted(bounds, rank + 0.5) in the reference).
// ---------------------------------------------------------------------------
__global__ void map_kernel(const float* __restrict__ sims, int N,
                           const int* __restrict__ ranks,
                           const float* __restrict__ means,
                           const float* __restrict__ bounds,
                           const int* __restrict__ nbs,
                           float* __restrict__ out2) {
  const int row = blockIdx.y;
  const int i = (int)(blockIdx.x * 256 + threadIdx.x);
  if (i >= N) return;
  const int nb = nbs[row];
  const float* bd = bounds + (long long)row * N;
  const float r = (float)ranks[(long long)row * N + i] + 0.5f;
  int lo = 0, hi = nb;
  while (lo < hi) {
    const int mid = (lo + hi) >> 1;
    if (bd[mid] > r) hi = mid; else lo = mid + 1;
  }
  const float dual = means[(long long)row * N + lo];
  const float zi = ZSCALE * sims[(long long)row * N + i];
  out2[(long long)row * N + i] = zi - dual;
}

// ---------------------------------------------------------------------------
extern "C" void kernel_launch(void* const* d_in, const int* in_sizes, int n_in,
                              void* d_out, int out_size, void* d_ws, size_t ws_size,
                              hipStream_t stream) {
  const float* q      = (const float*)d_in[0];   // (16, 384)
  const float* corpus = (const float*)d_in[1];   // (N, 384)
  const int N = in_sizes[1] / DIM;               // 50000

  float* sims = (float*)d_out;                   // output 0: (16, N)
  float* out2 = sims + (size_t)NQ * N;           // output 1: (16, N)

  int M = 1;                                     // pow2 pad for bitonic network
  while (M < N) M <<= 1;                         // 50000 -> 65536

  // workspace carve-up (chunk sizes keep every pointer 8B-aligned)
  char* ws = (char*)d_ws;
  float* qn      = (float*)ws;  ws += (size_t)NQ * DIM * sizeof(float);
  const size_t per = (size_t)NQ * N;
  int*   ranks   = (int*)ws;    ws += per * sizeof(int);
  float* ssorted = (float*)ws;  ws += per * sizeof(float);
  float* sums    = (float*)ws;  ws += per * sizeof(float);
  float* cnts    = (float*)ws;  ws += per * sizeof(float);
  unsigned long long* keys = (unsigned long long*)ws; ws += (size_t)NQ * M * sizeof(unsigned long long);
  int*   nbs     = (int*)ws;

  qnorm_kernel<<<NQ, 128, 0, stream>>>(q, qn);

  const int ntiles = (N + 15) / 16;
  const int gb = (ntiles + 7) / 8;               // 8 waves (256 thr) per block
  simgemm_kernel<<<gb, 256, 0, stream>>>(qn, corpus, sims, N);

  dim3 kg((unsigned)((M + 255) / 256), NQ);
  key_build_kernel<<<kg, 256, 0, stream>>>(sims, N, M, keys);

  // Bitonic sort: LDS chunk phase, then per merge level global substages + LDS tail.
  const int chnk = (M < CHNK) ? M : CHNK;
  dim3 cg((unsigned)(M / chnk), NQ);
  dim3 pg((unsigned)(((M >> 1) + 255) / 256), NQ);
  bitonic_chunk_kernel<<<cg, SORT_T, 0, stream>>>(keys, M, chnk);
  for (int k = chnk << 1; k <= M; k <<= 1) {
    for (int j = k >> 1; j >= chnk; j >>= 1)
      bitonic_global_kernel<<<pg, 256, 0, stream>>>(keys, M, k, j);
    bitonic_tail_kernel<<<cg, SORT_T, 0, stream>>>(keys, M, k);
  }

  dim3 ng((unsigned)((N + 255) / 256), NQ);
  unpack_kernel<<<ng, 256, 0, stream>>>(keys, sims, N, M, ranks, ssorted);

  pav_kernel<<<NQ, 32, 0, stream>>>(ssorted, N, sums, cnts, nbs);

  map_kernel<<<ng, 256, 0, stream>>>(sims, N, ranks, sums, cnts, nbs, out2);
}